// LeastSquares_53927609369084
// MI455X (gfx1250) — compile-verified
//
#include <hip/hip_runtime.h>
#include <stdint.h>

#define IW 1024
#define IH 1024
#define HWSZ (IW * IH)
#define TS 16
#define HALO 3
#define PADT (TS + 2 * HALO)   // 22
#define FPITCH PADT            // packed rows so TDM can DMA a whole plane
#define NMAPS 60               // 36 AtA upper-triangle + 24 AtY
#define CHUNK 20
#define EPSR 1e-4f

// ---------------- CDNA5 async global->LDS path ----------------
#if defined(__gfx1250__) && __has_builtin(__builtin_amdgcn_global_load_async_to_lds_b32)
#define HAVE_ASYNC 1
#else
#define HAVE_ASYNC 0
#endif

// ---------------- CDNA5 Tensor Data Mover path ----------------
#if defined(__gfx1250__) && __has_builtin(__builtin_amdgcn_tensor_load_to_lds)
#define HAVE_TDM 1
#else
#define HAVE_TDM 0
#endif

typedef __attribute__((address_space(1))) int* gptr_i32;
typedef __attribute__((address_space(3))) int* lptr_i32;
typedef unsigned int v4u __attribute__((ext_vector_type(4)));
typedef int v8i __attribute__((ext_vector_type(8)));
typedef int v4i __attribute__((ext_vector_type(4)));

__device__ __forceinline__ void async_ld_f32(const float* g, float* l) {
#if HAVE_ASYNC
  __builtin_amdgcn_global_load_async_to_lds_b32(
      (gptr_i32)(uintptr_t)g,
      (lptr_i32)(uint32_t)(uintptr_t)l,
      0, 0);
#else
  *l = *g;
#endif
}

__device__ __forceinline__ void async_wait_all() {
#if HAVE_ASYNC
#if __has_builtin(__builtin_amdgcn_s_wait_asynccnt)
  __builtin_amdgcn_s_wait_asynccnt(0);
#else
  asm volatile("s_wait_asynccnt 0x0" ::: "memory");
#endif
#endif
}

#if HAVE_TDM
// DMA one 22x22 f32 tile (rows packed) from global (row stride IW elements)
// into LDS at byte offset lds_off. D# packing per CDNA5 ISA ch.8.
__device__ __forceinline__ void tdm_load_tile(const float* gsrc, uint32_t lds_off) {
  const uint64_t ga = (uint64_t)(uintptr_t)gsrc;
  v4u g0;
  g0.x = 1u;                                    // count=1, user descriptor
  g0.y = lds_off;                               // lds_addr (bytes)
  g0.z = (uint32_t)(ga & 0xFFFFFFFFu);          // global_addr[31:0]
  g0.w = (uint32_t)((ga >> 32) & 0x01FFFFFFu)   // global_addr[56:32]
         | (2u << 30);                          // type=2 ("image")
  v8i g1;
  g1[0] = (int)(2u << 16);                      // wg_mask=0, data_size=2 (4B)
  g1[1] = (int)(((uint32_t)PADT & 0xFFFFu) << 16);       // tensor_dim0.lo16
  g1[2] = (int)(((uint32_t)PADT & 0xFFFFu) << 16);       // td0.hi=0 | tensor_dim1.lo16
  g1[3] = (int)(((uint32_t)PADT & 0xFFFFu) << 16);       // td1.hi=0 | tile_dim0=22
  g1[4] = (int)((uint32_t)PADT);                         // tile_dim1=22 | tile_dim2=0
  g1[5] = (int)IW;                                       // tensor_dim0_stride[31:0]
  g1[6] = 0;                                             // stride0.hi | stride1.lo
  g1[7] = 0;                                             // stride1.hi
  const v4i gz = {0, 0, 0, 0};                  // groups 2/3: 2D tensor
#if __clang_major__ >= 23
  // upstream clang-23 (amdgpu-toolchain): 6-arg form with trailing int32x8
  const v8i gpad = {0, 0, 0, 0, 0, 0, 0, 0};
  __builtin_amdgcn_tensor_load_to_lds(g0, g1, gz, gz, gpad, 0);
#else
  // ROCm 7.2 / AMD clang-22: 5-arg form
  __builtin_amdgcn_tensor_load_to_lds(g0, g1, gz, gz, 0);
#endif
}
#endif

// packed upper-triangle index for 8x8 symmetric matrix
__host__ __device__ constexpr int uidx(int i, int j) {
  return (i * (17 - i)) / 2 + (j - i);
}

// One chunk of CHUNK maps: horizontal 7-tap sums into LDS, barrier,
// per-pixel vertical 7-tap sums into register array s[].
template <int M0>
__device__ __forceinline__ void chunk_pass(const float (*feat)[PADT][FPITCH],
                                           float (*hsum)[PADT][TS],
                                           int tid, int tx, int ty,
                                           float s[NMAPS]) {
  for (int p = tid; p < PADT * TS; p += 256) {
    const int hy = p / TS;
    const int hx = p % TS;
    float acc[CHUNK];
#pragma unroll
    for (int mm = 0; mm < CHUNK; ++mm) acc[mm] = 0.0f;
#pragma unroll
    for (int dx = 0; dx < 7; ++dx) {
      float fv[8], yv[3];
#pragma unroll
      for (int ch = 0; ch < 8; ++ch) fv[ch] = feat[ch][hy][hx + dx];
#pragma unroll
      for (int c = 0; c < 3; ++c) yv[c] = feat[8 + c][hy][hx + dx];
#pragma unroll
      for (int i = 0; i < 8; ++i)
#pragma unroll
        for (int j = i; j < 8; ++j) {
          const int m = uidx(i, j);
          if (m >= M0 && m < M0 + CHUNK) acc[m - M0] += fv[i] * fv[j];
        }
#pragma unroll
      for (int i = 0; i < 8; ++i)
#pragma unroll
        for (int c = 0; c < 3; ++c) {
          const int m = 36 + i * 3 + c;
          if (m >= M0 && m < M0 + CHUNK) acc[m - M0] += fv[i] * yv[c];
        }
    }
#pragma unroll
    for (int mm = 0; mm < CHUNK; ++mm) hsum[mm][hy][hx] = acc[mm];
  }
  __syncthreads();
#pragma unroll
  for (int mm = 0; mm < CHUNK; ++mm) {
    float v = 0.0f;
#pragma unroll
    for (int dy = 0; dy < 7; ++dy) v += hsum[mm][ty + dy][tx];
    s[M0 + mm] = v;
  }
  __syncthreads();
}

__global__ __launch_bounds__(256) void LeastSquares_53927609369084_kernel(
    const float* __restrict__ input, const float* __restrict__ depth,
    const float* __restrict__ albedo, const float* __restrict__ normal,
    float* __restrict__ out) {
  __shared__ float feat[11][PADT][FPITCH];  // 0=ones,1=depth,2-4=albedo,5-7=normal,8-10=y
  __shared__ float hsum[CHUNK][PADT][TS];

  const int tid = threadIdx.x;
  const int tx = tid & 15;
  const int ty = tid >> 4;
  const int bx = blockIdx.x, by = blockIdx.y;
  const int gx0 = bx * TS - HALO;
  const int gy0 = by * TS - HALO;

  const float* src[10] = {depth,
                          albedo, albedo + HWSZ, albedo + 2 * HWSZ,
                          normal, normal + HWSZ, normal + 2 * HWSZ,
                          input,  input + HWSZ,  input + 2 * HWSZ};

  const bool interior =
      (gx0 >= 0) && (gy0 >= 0) && (gx0 + PADT <= IW) && (gy0 + PADT <= IH);

  if (interior) {
#if HAVE_TDM
    // One TDM descriptor per feature plane, issued by wave 0 only.
    const int base_off = gy0 * IW + gx0;
    if (tid < 32) {
#pragma unroll
      for (int k = 0; k < 10; ++k)
        tdm_load_tile(src[k] + base_off,
                      (uint32_t)(uintptr_t)(&feat[k + 1][0][0]));
      __builtin_amdgcn_s_wait_tensorcnt(0);
    }
    for (int p = tid; p < PADT * PADT; p += 256)
      feat[0][p / PADT][p % PADT] = 1.0f;
#else
    for (int p = tid; p < PADT * PADT; p += 256) {
      const int ly = p / PADT;
      const int lx = p % PADT;
      const int goff = (gy0 + ly) * IW + (gx0 + lx);
      feat[0][ly][lx] = 1.0f;
#pragma unroll
      for (int k = 0; k < 10; ++k) async_ld_f32(src[k] + goff, &feat[k + 1][ly][lx]);
    }
    async_wait_all();
#endif
  } else {
    // zero-fill outside the image: exactly reproduces the clipped window sums
    for (int p = tid; p < PADT * PADT; p += 256) {
      const int ly = p / PADT;
      const int lx = p % PADT;
      const int gy = gy0 + ly, gx = gx0 + lx;
      const bool in = (gx >= 0) & (gx < IW) & (gy >= 0) & (gy < IH);
      const int goff = in ? gy * IW + gx : 0;
      feat[0][ly][lx] = in ? 1.0f : 0.0f;
#pragma unroll
      for (int k = 0; k < 10; ++k) feat[k + 1][ly][lx] = in ? src[k][goff] : 0.0f;
    }
  }
  __syncthreads();

  float s[NMAPS];
  chunk_pass<0>(feat, hsum, tid, tx, ty, s);
  chunk_pass<20>(feat, hsum, tid, tx, ty, s);
  chunk_pass<40>(feat, hsum, tid, tx, ty, s);

  // ---- per-pixel symmetric 8x8 solve, 3 RHS (SPD + eps -> no pivoting) ----
  float a[36];
#pragma unroll
  for (int m = 0; m < 36; ++m) a[m] = s[m];
#pragma unroll
  for (int i = 0; i < 8; ++i) a[uidx(i, i)] += EPSR;

  float b[8][3];
#pragma unroll
  for (int i = 0; i < 8; ++i)
#pragma unroll
    for (int c = 0; c < 3; ++c) b[i][c] = s[36 + i * 3 + c];

  float inv[8];
#pragma unroll
  for (int k = 0; k < 8; ++k) {
    inv[k] = 1.0f / a[uidx(k, k)];
#pragma unroll
    for (int i = k + 1; i < 8; ++i) {
      const float f = a[uidx(k, i)] * inv[k];
#pragma unroll
      for (int j = i; j < 8; ++j) a[uidx(i, j)] -= f * a[uidx(k, j)];
#pragma unroll
      for (int c = 0; c < 3; ++c) b[i][c] -= f * b[k][c];
    }
  }
#pragma unroll
  for (int i = 7; i >= 0; --i)
#pragma unroll
    for (int c = 0; c < 3; ++c) {
      float x = b[i][c];
#pragma unroll
      for (int j = i + 1; j < 8; ++j) x -= a[uidx(i, j)] * b[j][c];
      b[i][c] = x * inv[i];
    }

  // evaluate fitted model at the center pixel's features
  float fc[8];
#pragma unroll
  for (int ch = 0; ch < 8; ++ch) fc[ch] = feat[ch][ty + HALO][tx + HALO];

  const int gy = by * TS + ty;
  const int gx = bx * TS + tx;
#pragma unroll
  for (int c = 0; c < 3; ++c) {
    float o = 0.0f;
#pragma unroll
    for (int i = 0; i < 8; ++i) o += b[i][c] * fc[i];
    out[c * HWSZ + gy * IW + gx] = o;
  }
}

extern "C" void kernel_launch(void* const* d_in, const int* in_sizes, int n_in,
                              void* d_out, int out_size, void* d_ws, size_t ws_size,
                              hipStream_t stream) {
  const float* input  = (const float*)d_in[0];  // [1,3,1024,1024]
  const float* depth  = (const float*)d_in[1];  // [1,1,1024,1024]
  const float* albedo = (const float*)d_in[2];  // [1,3,1024,1024]
  const float* normal = (const float*)d_in[3];  // [1,3,1024,1024]
  float* out = (float*)d_out;                   // [1,3,1024,1024]

  dim3 grid(IW / TS, IH / TS);
  dim3 block(256);
  hipLaunchKernelGGL(LeastSquares_53927609369084_kernel, grid, block, 0, stream,
                     input, depth, albedo, normal, out);
}